// RotaryPositionalEmbedding_64433099374867
// MI455X (gfx1250) — compile-verified
//
#include <hip/hip_runtime.h>
#include <hip/hip_bf16.h>
#include <stdint.h>

#define SEQ  2048
#define DK   128
#define BHN  64      // BATCH * HEADS = 4 * 16

typedef __attribute__((ext_vector_type(4))) float v4f;

// One block per output row s'. 256 threads = 8 waves.
// Lanes 0..63 DMA the 64 (cos,sin) pairs of rope[p]'s 2x2-block diagonal into
// LDS with CDNA5 async-tensor loads (ASYNCcnt path), then 8 groups of 32
// lanes stream the 64 (b,h) rows with B128 loads and non-temporal B128 stores.
__global__ __launch_bounds__(256) void rope_gather_rotate_kernel(
    const float* __restrict__ x,      // [4,16,2048,128]
    const float* __restrict__ rope,   // [4096,128,128]
    const int*   __restrict__ tpos,   // [2048]
    float*       __restrict__ out)    // [4,16,2048,128]
{
    __shared__ __attribute__((aligned(16))) float cs[DK]; // c0,s0,c1,s1,...

    const int sp  = blockIdx.x;        // output row
    const int p   = tpos[sp];          // gathered source row / rope position
    const int tid = threadIdx.x;

    // ---- Stage cos/sin via async DMA to LDS (gfx1250: GLOBAL_LOAD_ASYNC_TO_LDS_B64)
    // rope[p, 2i, 2i] = cos_i, rope[p, 2i, 2i+1] = sin_i  -> float2 at byte 1032*i
    if (tid < 64) {
        const float* src = rope + (size_t)p * (DK * DK);
        uint32_t lds_addr = (uint32_t)(uintptr_t)(&cs[2 * tid]); // LDS byte addr (low 32 bits)
        uint32_t voff     = (uint32_t)(tid * ((2 * DK + 2) * 4)); // 1032 * i bytes
        asm volatile("global_load_async_to_lds_b64 %0, %1, %2"
                     :
                     : "v"(lds_addr), "v"(voff), "s"(src)
                     : "memory");
    }
    asm volatile("s_wait_asynccnt 0" ::: "memory");
    __syncthreads();

    const int grp  = tid >> 5;   // 0..7: which (b,h) row this group handles
    const int lane = tid & 31;   // 0..31 within the row
    const int base = lane * 4;   // float offset within the 128-wide row

    // (cos_j, sin_j, cos_j+1, sin_j+1) — LDS layout matches a straight float4
    const v4f cv = *reinterpret_cast<const v4f*>(&cs[base]);

    for (int bh = grp; bh < BHN; bh += 8) {
        const float* xrow = x + ((size_t)bh * SEQ + p) * DK;
        if (bh + 8 < BHN) {
            // gfx1250 global_prefetch_b8 for the next row this group will touch
            __builtin_prefetch(x + ((size_t)(bh + 8) * SEQ + p) * DK + base, 0, 0);
        }
        v4f xv = *reinterpret_cast<const v4f*>(xrow + base);
        v4f ov;
        ov.x = xv.x * cv.x - xv.y * cv.y;
        ov.y = xv.x * cv.y + xv.y * cv.x;
        ov.z = xv.z * cv.z - xv.w * cv.w;
        ov.w = xv.z * cv.w + xv.w * cv.z;
        // Output is never re-read: non-temporal store keeps x/rope resident in L2
        __builtin_nontemporal_store(
            ov, reinterpret_cast<v4f*>(out + ((size_t)bh * SEQ + sp) * DK + base));
    }
}

extern "C" void kernel_launch(void* const* d_in, const int* in_sizes, int n_in,
                              void* d_out, int out_size, void* d_ws, size_t ws_size,
                              hipStream_t stream)
{
    const float* x    = (const float*)d_in[0];   // [4,16,2048,128] f32
    const float* rope = (const float*)d_in[1];   // [4096,128,128]  f32
    const int*   tp   = (const int*)d_in[2];     // [2048] i32
    float*       out  = (float*)d_out;           // [4,16,2048,128] f32

    rope_gather_rotate_kernel<<<SEQ, 256, 0, stream>>>(x, rope, tp, out);
}